// GCN_29411936043071
// MI455X (gfx1250) — compile-verified
//
#include <hip/hip_runtime.h>

// ---------------------------------------------------------------------------
// GCN (2x GCNConv + mean-pool + linear) for MI455X / gfx1250.
// GEMMs use V_WMMA_F32_16X16X4_F32 (fp32 matrix op -> matches fp32 reference).
// W is staged into LDS with GLOBAL_LOAD_ASYNC_TO_LDS_B128 (ASYNCcnt path).
// Sparse aggregation uses global_atomic_add_f32; edge working set is L2-resident.
// ---------------------------------------------------------------------------

typedef __attribute__((ext_vector_type(2))) float v2f;
typedef __attribute__((ext_vector_type(8))) float v8f;

#define FDIM 128

// ---------------------------- small utility kernels ------------------------

__global__ void init_const_kernel(float* __restrict__ p, float v, int n) {
    int i = blockIdx.x * blockDim.x + threadIdx.x;
    if (i < n) p[i] = v;
}

__global__ void count_deg_kernel(const int* __restrict__ dst, float* __restrict__ deg, int nE) {
    int e = blockIdx.x * blockDim.x + threadIdx.x;
    if (e < nE) atomicAdd(&deg[dst[e]], 1.0f);
}

__global__ void rsqrt_kernel(float* __restrict__ deg, int n) {
    int i = blockIdx.x * blockDim.x + threadIdx.x;
    if (i < n) deg[i] = rsqrtf(fmaxf(deg[i], 1.0f));
}

// agg[i][f] = h[i][f] * dinv[i]^2   (self-loop contribution; also zero-seeds agg)
__global__ void selfloop_seed_kernel(const float* __restrict__ h,
                                     const float* __restrict__ dinv,
                                     float* __restrict__ agg, int n) {
    long long i = (long long)blockIdx.x * blockDim.x + threadIdx.x;
    long long tot = (long long)n * FDIM;
    if (i < tot) {
        int node = (int)(i >> 7);
        float w = dinv[node];
        agg[i] = h[i] * w * w;
    }
}

// one wave per edge: lane l handles features 4l..4l+3
__global__ void scatter_edges_kernel(const float* __restrict__ h,
                                     const int* __restrict__ src,
                                     const int* __restrict__ dst,
                                     const float* __restrict__ dinv,
                                     float* __restrict__ agg, int nE) {
    int e = blockIdx.x * (blockDim.x >> 5) + (threadIdx.x >> 5);
    if (e >= nE) return;
    int lane = threadIdx.x & 31;
    int s = src[e];
    int d = dst[e];
    float w = dinv[s] * dinv[d];
    const float4* hs = (const float4*)(h + (long long)s * FDIM);
    float4 v = hs[lane];
    float* o = agg + (long long)d * FDIM + lane * 4;
    atomicAdd(&o[0], v.x * w);
    atomicAdd(&o[1], v.y * w);
    atomicAdd(&o[2], v.z * w);
    atomicAdd(&o[3], v.w * w);
}

__global__ void bias_relu_kernel(float* __restrict__ h, const float* __restrict__ b, int n) {
    long long i = (long long)blockIdx.x * blockDim.x + threadIdx.x;
    long long tot = (long long)n * FDIM;
    if (i < tot) h[i] = fmaxf(h[i] + b[i & (FDIM - 1)], 0.0f);
}

// one wave per node: accumulate per-graph feature sums + node counts
__global__ void pool_kernel(const float* __restrict__ h, const int* __restrict__ batch,
                            float* __restrict__ sums, float* __restrict__ cnt, int n) {
    int node = blockIdx.x * (blockDim.x >> 5) + (threadIdx.x >> 5);
    if (node >= n) return;
    int lane = threadIdx.x & 31;
    int g = batch[node];
    const float4* hp = (const float4*)(h + (long long)node * FDIM);
    float4 v = hp[lane];
    float* s = sums + (long long)g * FDIM + lane * 4;
    atomicAdd(&s[0], v.x);
    atomicAdd(&s[1], v.y);
    atomicAdd(&s[2], v.z);
    atomicAdd(&s[3], v.w);
    if (lane == 0) atomicAdd(&cnt[g], 1.0f);
}

// out[g] = (sum_f sums[g][f] * Wlin[f]) / max(cnt[g],1) + blin
__global__ void finalize_kernel(const float* __restrict__ sums, const float* __restrict__ cnt,
                                const float* __restrict__ Wlin, const float* __restrict__ blin,
                                float* __restrict__ out, int nGraphs) {
    int g = blockIdx.x * blockDim.x + threadIdx.x;
    if (g >= nGraphs) return;
    float acc = 0.0f;
    const float* s = sums + (long long)g * FDIM;
#pragma unroll 8
    for (int f = 0; f < FDIM; ++f) acc += s[f] * Wlin[f];
    out[g] = acc / fmaxf(cnt[g], 1.0f) + blin[0];
}

// ------------------------------- WMMA GEMM ---------------------------------
// C[M,128] = A[M,128] @ W[128,128]   (no bias / no activation here)
// 256 threads = 8 waves; block strip = 128 rows; W staged in 64KB LDS via the
// CDNA5 async global->LDS path (no VGPR round-trip, tracked by ASYNCcnt).
// Each wave: 16 rows, 8 N-tiles, K loop of 32 steps of K=4 fp32 WMMA.
__global__ __launch_bounds__(256) void gemm128_wmma_kernel(const float* __restrict__ A,
                                                           const float* __restrict__ W,
                                                           float* __restrict__ C, int M) {
    __shared__ float sW[FDIM * FDIM];  // 64 KB of the 320 KB/WGP LDS

    const int tid = threadIdx.x;

    // Async bulk copy W (128x128 fp32 = 4096 x 16B) into LDS:
    // per-lane LDS byte offset in one VGPR, 64-bit global address in a pair.
    // LDS generic addresses are {SHARED_BASE, offset32} -> low 32 bits are the
    // LDS byte offset.
    {
        unsigned int ldsBase = (unsigned int)(unsigned long long)(uintptr_t)sW;
        const char* gW = (const char*)W;
#pragma unroll
        for (int i = tid; i < (FDIM * FDIM) / 4; i += 256) {
            unsigned int lo = ldsBase + (unsigned int)i * 16u;
            unsigned long long ga = (unsigned long long)(uintptr_t)(gW + (long long)i * 16);
            asm volatile("global_load_async_to_lds_b128 %0, %1, off"
                         :: "v"(lo), "v"(ga)
                         : "memory");
        }
        asm volatile("s_wait_asynccnt 0" ::: "memory");
    }
    __syncthreads();

    const int wave  = tid >> 5;          // 0..7
    const int lane  = tid & 31;
    const int lhalf = lane & 15;
    const bool hi   = lane >= 16;        // lanes 16..31 carry K+2/K+3 (A) rows K=2,3 (B)
    const int kofs  = hi ? 2 : 0;

    const int r0   = blockIdx.x * 128 + wave * 16;  // first row of this wave's strip
    const int arow = r0 + lhalf;
    const float* aptr = A + (long long)((arow < M) ? arow : 0) * FDIM;  // clamp OOB loads,
                                                                        // keep EXEC all-ones
    v8f acc[8];
#pragma unroll
    for (int j = 0; j < 8; ++j) acc[j] = (v8f){};

#pragma unroll 4
    for (int k0 = 0; k0 < FDIM; k0 += 4) {
        // A fragment (16x4 fp32): lane half picks K pair {k0,k0+1} or {k0+2,k0+3}
        v2f a;
        a.x = aptr[k0 + kofs];
        a.y = aptr[k0 + kofs + 1];
#pragma unroll
        for (int j = 0; j < 8; ++j) {
            // B fragment (4x16 fp32): rows K=k0+kofs, k0+kofs+1; cols j*16+lhalf
            v2f b;
            b.x = sW[(k0 + kofs) * FDIM + j * 16 + lhalf];
            b.y = sW[(k0 + kofs + 1) * FDIM + j * 16 + lhalf];
            acc[j] = __builtin_amdgcn_wmma_f32_16x16x4_f32(
                false, a, false, b, (short)0, acc[j], false, false);
        }
    }

    // C/D layout: VGPR v -> row (hi ? 8+v : v), col = j*16 + lhalf
#pragma unroll
    for (int j = 0; j < 8; ++j) {
#pragma unroll
        for (int v = 0; v < 8; ++v) {
            int row = r0 + (hi ? 8 + v : v);
            if (row < M) C[(long long)row * FDIM + j * 16 + lhalf] = acc[j][v];
        }
    }
}

// ------------------------------- launcher ----------------------------------

extern "C" void kernel_launch(void* const* d_in, const int* in_sizes, int n_in,
                              void* d_out, int out_size, void* d_ws, size_t ws_size,
                              hipStream_t stream) {
    const float* x     = (const float*)d_in[0];
    const int*   eidx  = (const int*)d_in[1];     // [2, nE] flat: row0=src, row1=dst
    const int*   batch = (const int*)d_in[2];
    const float* W1    = (const float*)d_in[3];
    const float* b1    = (const float*)d_in[4];
    const float* W2    = (const float*)d_in[5];
    const float* b2    = (const float*)d_in[6];
    const float* Wlin  = (const float*)d_in[7];
    const float* blin  = (const float*)d_in[8];
    float* out = (float*)d_out;

    const int nNodes  = in_sizes[0] / FDIM;
    const int nE      = in_sizes[1] / 2;
    const int nGraphs = out_size;
    const int* src = eidx;
    const int* dst = eidx + nE;

    // workspace carve-up
    char* ws = (char*)d_ws;
    const long long featBytes = (long long)nNodes * FDIM * sizeof(float);
    float* bufA = (float*)ws;                                  // h = A@W
    float* bufB = (float*)(ws + featBytes);                    // aggregated / activated
    float* deg  = (float*)(ws + 2 * featBytes);                // deg -> dinv (in place)
    float* sums = (float*)(ws + 2 * featBytes + ((long long)nNodes * sizeof(float) + 255 & ~255LL));
    float* cnt  = sums + (long long)nGraphs * FDIM;
    (void)ws_size; (void)n_in;

    const int T = 256;
    const int featBlocks = (int)(((long long)nNodes * FDIM + T - 1) / T);
    const int edgeWaveBlocks = (nE + (T / 32) - 1) / (T / 32);
    const int nodeWaveBlocks = (nNodes + (T / 32) - 1) / (T / 32);
    const int gemmBlocks = (nNodes + 127) / 128;

    // ---- degrees (self-loop => start at 1.0), then dinv = rsqrt(deg) ----
    init_const_kernel<<<(nNodes + T - 1) / T, T, 0, stream>>>(deg, 1.0f, nNodes);
    count_deg_kernel<<<(nE + T - 1) / T, T, 0, stream>>>(dst, deg, nE);
    rsqrt_kernel<<<(nNodes + T - 1) / T, T, 0, stream>>>(deg, nNodes);

    // ---- layer 1 ----
    gemm128_wmma_kernel<<<gemmBlocks, T, 0, stream>>>(x, W1, bufA, nNodes);
    selfloop_seed_kernel<<<featBlocks, T, 0, stream>>>(bufA, deg, bufB, nNodes);
    scatter_edges_kernel<<<edgeWaveBlocks, T, 0, stream>>>(bufA, src, dst, deg, bufB, nE);
    bias_relu_kernel<<<featBlocks, T, 0, stream>>>(bufB, b1, nNodes);

    // ---- layer 2 ----
    gemm128_wmma_kernel<<<gemmBlocks, T, 0, stream>>>(bufB, W2, bufA, nNodes);
    selfloop_seed_kernel<<<featBlocks, T, 0, stream>>>(bufA, deg, bufB, nNodes);
    scatter_edges_kernel<<<edgeWaveBlocks, T, 0, stream>>>(bufA, src, dst, deg, bufB, nE);
    bias_relu_kernel<<<featBlocks, T, 0, stream>>>(bufB, b2, nNodes);

    // ---- mean pool + linear head ----
    init_const_kernel<<<(nGraphs * FDIM + T - 1) / T, T, 0, stream>>>(sums, 0.0f, nGraphs * FDIM);
    init_const_kernel<<<1, T, 0, stream>>>(cnt, 0.0f, nGraphs);
    pool_kernel<<<nodeWaveBlocks, T, 0, stream>>>(bufB, batch, sums, cnt, nNodes);
    finalize_kernel<<<(nGraphs + T - 1) / T, T, 0, stream>>>(sums, cnt, Wlin, blin, out, nGraphs);
}